// MHCResidualMixer_6339371729051
// MI455X (gfx1250) — compile-verified
//
#include <hip/hip_runtime.h>
#include <hip/hip_bf16.h>

// ---------------------------------------------------------------------------
// Stage 1: Sinkhorn on the 8x8 logits (20 iters). ~3 KFLOP total -> a single
// thread is plenty; runtime is dominated by the streaming mixer (bandwidth
// bound: 512 MB / 23.3 TB/s ~= 22 us). H -> d_ws as row-major [8][8] f32.
// ---------------------------------------------------------------------------
__global__ void sinkhorn8_kernel(const float* __restrict__ logits,
                                 float* __restrict__ H) {
    if (threadIdx.x != 0 || blockIdx.x != 0) return;
    const float eps = 1e-8f;
    float p[64];
    // temp = max(1.0, 1e-6) == 1.0, so logits/temp == logits
    for (int i = 0; i < 8; ++i) {
        float mx = logits[i * 8 + 0];
        for (int j = 1; j < 8; ++j) mx = fmaxf(mx, logits[i * 8 + j]);
        for (int j = 0; j < 8; ++j)
            p[i * 8 + j] = expf(logits[i * 8 + j] - mx) + eps;
    }
#pragma unroll 1
    for (int it = 0; it < 20; ++it) {
        for (int i = 0; i < 8; ++i) {           // row normalize (axis=-1)
            float s = 0.f;
            for (int j = 0; j < 8; ++j) s += p[i * 8 + j];
            const float d = s + eps;
            for (int j = 0; j < 8; ++j) p[i * 8 + j] /= d;
        }
        for (int j = 0; j < 8; ++j) {           // col normalize (axis=-2)
            float s = 0.f;
            for (int i = 0; i < 8; ++i) s += p[i * 8 + j];
            const float d = s + eps;
            for (int i = 0; i < 8; ++i) p[i * 8 + j] /= d;
        }
    }
    for (int k = 0; k < 64; ++k) H[k] = p[k];
}

// ---------------------------------------------------------------------------
// Stage 2: out[b,i,p] = sum_j H[i,j] * streams[b,j,p],  p in [0, P=T*D)
//
// One wave handles 16 consecutive positions per iteration via
// V_WMMA_F32_16X16X4_F32 (full fp32): D(16x16) += A(16x4)*B(4x16), chained
// twice to cover K = 8 streams.
//   A = H (rows 8..15 zero), loop invariant.
//   B = 4 stream rows x 16 positions; each B-VGPR load = two contiguous
//       64B segments (coalesced b32), non-temporal (single-touch stream,
//       512 MB > 192 MB L2 -> don't pollute).
//   D rows 0..7 (lanes 0..15 of acc regs 0..7) -> eight contiguous 64B
//       non-temporal stores.
// All addressing: uniform SGPR batch base + 32-bit unsigned lane offsets
// advanced by a uniform stride -> saddr+voffset form, minimal VALU overhead.
// EXEC is all-ones at every WMMA (wave-uniform loop; only stores predicated).
// ---------------------------------------------------------------------------
typedef __attribute__((ext_vector_type(2))) float v2f;
typedef __attribute__((ext_vector_type(8))) float v8f;

__global__ void mix8_wmma_kernel(const float* __restrict__ streams,
                                 const float* __restrict__ H,
                                 float* __restrict__ out,
                                 unsigned P) {               // T*D (elements)
    const int lane = threadIdx.x & 31;
    const int m    = lane & 15;     // A row / B,D column within tile
    const int hi   = lane >> 4;     // selects K rows {0,1} vs {2,3}

    // A matrix (H) in WMMA 16x4 f32 layout for the two K-halves.
    // v[0]: K = 2*hi, v[1]: K = 2*hi+1 (ISA 16x4 32-bit A layout).
    v2f aLo = {0.f, 0.f};
    v2f aHi = {0.f, 0.f};
    if (m < 8) {
        aLo.x = H[m * 8 + 2 * hi + 0];
        aLo.y = H[m * 8 + 2 * hi + 1];
        aHi.x = H[m * 8 + 2 * hi + 4];
        aHi.y = H[m * 8 + 2 * hi + 5];
    }

    // Per-lane 32-bit element offsets (max 8*P < 2^31), loop-invariant.
    const unsigned j0  = 2u * hi;                 // B v[0] stream row
    const unsigned j1  = 2u * hi + 1u;            // B v[1] stream row
    const unsigned lo0 = (j0 + 0u) * P + (unsigned)m;
    const unsigned lo1 = (j1 + 0u) * P + (unsigned)m;
    const unsigned lo2 = (j0 + 4u) * P + (unsigned)m;
    const unsigned lo3 = (j1 + 4u) * P + (unsigned)m;
    unsigned so[8];
#pragma unroll
    for (int r = 0; r < 8; ++r) so[r] = (unsigned)r * P + (unsigned)m;

    const unsigned waveId   = (blockIdx.x * blockDim.x + threadIdx.x) >> 5;
    const unsigned numWaves = (gridDim.x * blockDim.x) >> 5;
    const unsigned step     = numWaves * 16u;     // positions per grid-stride

    for (int b = 0; b < 4; ++b) {                 // uniform batch loop
        const float* __restrict__ sb = streams + (size_t)b * 8u * P;
        float*       __restrict__ ob = out     + (size_t)b * 8u * P;
#pragma unroll 2
        for (unsigned p0 = waveId * 16u; p0 < P; p0 += step) {
            v2f bLo, bHi;
            bLo.x = __builtin_nontemporal_load(sb + (p0 + lo0));
            bLo.y = __builtin_nontemporal_load(sb + (p0 + lo1));
            bHi.x = __builtin_nontemporal_load(sb + (p0 + lo2));
            bHi.y = __builtin_nontemporal_load(sb + (p0 + lo3));

            v8f c = {};
            // 8 args: (neg_a, A, neg_b, B, c_mod, C, reuse_a, reuse_b)
            c = __builtin_amdgcn_wmma_f32_16x16x4_f32(
                false, aLo, false, bLo, (short)0, c, false, false);
            c = __builtin_amdgcn_wmma_f32_16x16x4_f32(
                false, aHi, false, bHi, (short)0, c, false, false);

            // D rows 0..7 live in lanes 0..15 (rows 8..15 are zero).
            if (lane < 16) {
#pragma unroll
                for (int r = 0; r < 8; ++r)
                    __builtin_nontemporal_store(c[r], ob + (p0 + so[r]));
            }
        }
    }
}

extern "C" void kernel_launch(void* const* d_in, const int* in_sizes, int n_in,
                              void* d_out, int out_size, void* d_ws, size_t ws_size,
                              hipStream_t stream) {
    const float* streams = (const float*)d_in[0];   // [B=4, N=8, T=2048, D=1024] f32
    const float* logits  = (const float*)d_in[1];   // [8, 8] f32
    float*       out     = (float*)d_out;           // [B, N, T, D] f32
    float*       H       = (float*)d_ws;            // 64 floats scratch

    // Stage 1: tiny Sinkhorn (stream-ordered before the mixer).
    sinkhorn8_kernel<<<1, 32, 0, stream>>>(logits, H);

    // Stage 2: bandwidth-bound mix via f32 WMMA.
    const long long total = (long long)in_sizes[0]; // B*N*T*D
    const int B = 4, N = 8;
    const unsigned P = (unsigned)(total / (B * N)); // T*D = 2M

    const int threads = 256;                        // 8 wave32s per block
    const int blocks  = 2048;                       // 16384 waves; 131072
                                                    // tiles/batch -> 8 per wave
    mix8_wmma_kernel<<<blocks, threads, 0, stream>>>(streams, H, out, P);
}